// HypAgg_55516747268124
// MI455X (gfx1250) — compile-verified
//
#include <hip/hip_runtime.h>

#define FEAT 128
#define WV 8   // waves (rows) per 256-thread block

#if defined(__HIP_DEVICE_COMPILE__) && \
    __has_builtin(__builtin_amdgcn_global_load_async_to_lds_b64) && \
    __has_builtin(__builtin_amdgcn_s_wait_asynccnt)
#define HAVE_ASYNC 1
typedef int v2i_async __attribute__((ext_vector_type(2)));
typedef __attribute__((address_space(1))) v2i_async* as1_v2i_p;  // global (AS1)
typedef __attribute__((address_space(3))) v2i_async* as3_v2i_p;  // LDS (AS3)
#else
#define HAVE_ASYNC 0
#endif

// ---------------------------------------------------------------------------
// Shared epilogue: expmap0 (c=1) + Poincare-ball proj, one wave owns one row.
// ---------------------------------------------------------------------------
__device__ __forceinline__ void finalize_store(float4 a, float* __restrict__ out,
                                               int r, int lane) {
    float ss = a.x * a.x + a.y * a.y + a.z * a.z + a.w * a.w;
#pragma unroll
    for (int off = 16; off > 0; off >>= 1)
        ss += __shfl_xor(ss, off, 32);           // wave32 butterfly reduction
    const float un = fmaxf(sqrtf(ss), 1e-15f);   // u_norm clamped at MIN_NORM
    const float th = tanhf(un);                  // sqrt_c == 1
    const float f  = th / un;
    a.x *= f; a.y *= f; a.z *= f; a.w *= f;
    // ||p|| == th (up to fp noise); proj clamp to (1 - BALL_EPS)
    const float pn = fmaxf(th, 1e-15f);
    const float maxnorm = 1.0f - 4e-3f;
    if (pn > maxnorm) {
        const float g = maxnorm / pn;
        a.x *= g; a.y *= g; a.z *= g; a.w *= g;
    }
    *((float4*)(out + (size_t)r * FEAT) + lane) = a;
}

// ---------------------------------------------------------------------------
// 1) per-row non-self-loop edge counts
// ---------------------------------------------------------------------------
__global__ __launch_bounds__(256) void count_kernel(const int* __restrict__ row,
                                                    const int* __restrict__ col,
                                                    unsigned* __restrict__ cnt, int E) {
    const int e = blockIdx.x * 256 + threadIdx.x;
    if (e < E) {
        const int r = row[e];
        if (r != col[e]) atomicAdd(&cnt[r], 1u);
    }
}

// ---------------------------------------------------------------------------
// 2) single-block exclusive scan over N counts -> offs[0..N]
// ---------------------------------------------------------------------------
__global__ __launch_bounds__(1024) void scan_kernel(const unsigned* __restrict__ cnt,
                                                    unsigned* __restrict__ offs, int n) {
    __shared__ unsigned sd[1024];
    __shared__ unsigned carry_s;
    if (threadIdx.x == 0) carry_s = 0u;
    __syncthreads();
    for (int base = 0; base < n; base += 1024) {
        const int i = base + (int)threadIdx.x;
        const unsigned v = (i < n) ? cnt[i] : 0u;
        sd[threadIdx.x] = v;
        __syncthreads();
        for (int off = 1; off < 1024; off <<= 1) {
            const unsigned t = (threadIdx.x >= (unsigned)off) ? sd[threadIdx.x - off] : 0u;
            __syncthreads();
            sd[threadIdx.x] += t;
            __syncthreads();
        }
        if (i < n) offs[i] = carry_s + sd[threadIdx.x] - v;   // exclusive
        __syncthreads();
        if (threadIdx.x == 1023) carry_s += sd[1023];
        __syncthreads();
    }
    if (threadIdx.x == 0) offs[n] = carry_s;
}

// ---------------------------------------------------------------------------
// 3) dis = deg^{-1/2}, deg = cnt + 1 (appended self-loop) >= 1 always
// ---------------------------------------------------------------------------
__global__ __launch_bounds__(256) void dis_kernel(const unsigned* __restrict__ cnt,
                                                  float* __restrict__ dis, int n) {
    const int i = blockIdx.x * 256 + threadIdx.x;
    if (i < n) dis[i] = 1.0f / sqrtf((float)(cnt[i] + 1u));
}

// ---------------------------------------------------------------------------
// 4) CSR fill: entry = {col, precomputed weight dis[r]*dis[c]}
// ---------------------------------------------------------------------------
__global__ __launch_bounds__(256) void fill_kernel(const int* __restrict__ row,
                                                   const int* __restrict__ col,
                                                   const unsigned* __restrict__ offs,
                                                   unsigned* __restrict__ cur,
                                                   const float* __restrict__ dis,
                                                   uint2* __restrict__ csr, int E) {
    const int e = blockIdx.x * 256 + threadIdx.x;
    if (e >= E) return;
    const int r = row[e], c = col[e];
    if (r == c) return;                      // dropped self-loops (weight 0)
    const unsigned p = offs[r] + atomicAdd(&cur[r], 1u);
    csr[p] = make_uint2((unsigned)c, __float_as_uint(dis[r] * dis[c]));
}

// ---------------------------------------------------------------------------
// 5) wave-per-row aggregation with async LDS staging of neighbor lists,
//    fused hyperbolic epilogue.
// ---------------------------------------------------------------------------
__global__ __launch_bounds__(256) void agg_kernel(const float* __restrict__ x,
                                                  const uint2* __restrict__ csr,
                                                  const unsigned* __restrict__ offs,
                                                  const float* __restrict__ dis,
                                                  float* __restrict__ out, int N) {
    __shared__ uint2 stage[WV][32];
    const int wv   = (int)(threadIdx.x >> 5);
    const int lane = (int)(threadIdx.x & 31);
    const int r    = blockIdx.x * WV + wv;
    if (r >= N) return;

    // self-loop contribution: (dis[r]*1*dis[r]) * x[r]  ==  x[r]/deg[r]
    const float dr = dis[r];
    const float ws = dr * dr;
    const float4 xs = *((const float4*)(x + (size_t)r * FEAT) + lane);
    float4 acc = make_float4(ws * xs.x, ws * xs.y, ws * xs.z, ws * xs.w);

    const unsigned beg = offs[r], end = offs[r + 1];
    for (unsigned s = beg; s < end; s += 32u) {
        const unsigned rem = end - s;
        const int m = rem < 32u ? (int)rem : 32;
        if (lane < m) {
#if HAVE_ASYNC
            __builtin_amdgcn_global_load_async_to_lds_b64(
                (as1_v2i_p)(void*)(csr + s + lane),
                (as3_v2i_p)(&stage[wv][lane]), 0, 0);
#else
            stage[wv][lane] = csr[s + lane];
#endif
        }
#if HAVE_ASYNC
        __builtin_amdgcn_s_wait_asynccnt(0);
#else
        __threadfence_block();
#endif
        asm volatile("" ::: "memory");   // keep LDS reads after the wait
        for (int j = 0; j < m; ++j) {
            const uint2 e = stage[wv][j];                 // LDS broadcast read
            const float w = __uint_as_float(e.y);
            const float4 xv = *((const float4*)(x + (size_t)e.x * FEAT) + lane);
            acc.x = fmaf(w, xv.x, acc.x);
            acc.y = fmaf(w, xv.y, acc.y);
            acc.z = fmaf(w, xv.z, acc.z);
            acc.w = fmaf(w, xv.w, acc.w);
        }
        asm volatile("" ::: "memory");
    }
    finalize_store(acc, out, r, lane);
}

// ---------------------------------------------------------------------------
// Fallback path (small workspace): atomic scatter into d_out.
// ---------------------------------------------------------------------------
__global__ __launch_bounds__(256) void scatter_kernel(const int* __restrict__ row,
                                                      const int* __restrict__ col,
                                                      const float* __restrict__ dis,
                                                      const float* __restrict__ x,
                                                      float* __restrict__ out, int E) {
    const int gid  = blockIdx.x * 256 + threadIdx.x;
    const int e    = gid >> 5;
    const int lane = gid & 31;
    if (e >= E) return;
    const int r = row[e], c = col[e];
    if (r == c) return;
    const float w = dis[r] * dis[c];
    const float4 xv = *((const float4*)(x + (size_t)c * FEAT) + lane);
    float* o = out + (size_t)r * FEAT + lane * 4;
    atomicAdd(o + 0, w * xv.x);
    atomicAdd(o + 1, w * xv.y);
    atomicAdd(o + 2, w * xv.z);
    atomicAdd(o + 3, w * xv.w);
}

__global__ __launch_bounds__(256) void finalize_fb_kernel(const float* __restrict__ x,
                                                          const float* __restrict__ dis,
                                                          float* __restrict__ out, int N) {
    const int wv   = (int)(threadIdx.x >> 5);
    const int lane = (int)(threadIdx.x & 31);
    const int r    = blockIdx.x * WV + wv;
    if (r >= N) return;
    const float dr = dis[r];
    const float ws = dr * dr;
    const float4 xs = *((const float4*)(x + (size_t)r * FEAT) + lane);
    float4 a = *((const float4*)(out + (size_t)r * FEAT) + lane);
    a.x += ws * xs.x; a.y += ws * xs.y; a.z += ws * xs.z; a.w += ws * xs.w;
    finalize_store(a, out, r, lane);
}

// ---------------------------------------------------------------------------
extern "C" void kernel_launch(void* const* d_in, const int* in_sizes, int n_in,
                              void* d_out, int out_size, void* d_ws, size_t ws_size,
                              hipStream_t stream) {
    const float* x  = (const float*)d_in[0];
    const int*   ei = (const int*)d_in[1];
    const int N = in_sizes[0] / FEAT;
    const int E = in_sizes[1] / 2;
    const int* row = ei;
    const int* col = ei + E;
    float* out = (float*)d_out;

    char* base = (char*)d_ws;
    unsigned* cnt  = (unsigned*)base;                 // N
    unsigned* cur  = cnt + N;                         // N
    unsigned* offs = cur + N;                         // N+1
    float*    dis  = (float*)(offs + (size_t)N + 1);  // N
    size_t head = ((size_t)(4 * (size_t)N + 1) * 4 + 15) & ~(size_t)15;
    uint2* csr = (uint2*)(base + head);               // E entries of 8B
    const size_t need = head + (size_t)E * 8;

    const dim3 blk(256);
    const int eb = (E + 255) / 256;
    const int nb = (N + 255) / 256;
    const int rb = (N + WV - 1) / WV;

    if (ws_size >= need) {
        // CSR path (no float atomics in the hot loop)
        (void)hipMemsetAsync(cnt, 0, (size_t)2 * N * 4, stream);   // cnt + cur
        count_kernel<<<eb, blk, 0, stream>>>(row, col, cnt, E);
        scan_kernel<<<1, 1024, 0, stream>>>(cnt, offs, N);
        dis_kernel<<<nb, blk, 0, stream>>>(cnt, dis, N);
        fill_kernel<<<eb, blk, 0, stream>>>(row, col, offs, cur, dis, csr, E);
        agg_kernel<<<rb, blk, 0, stream>>>(x, csr, offs, dis, out, N);
    } else {
        // Minimal-workspace fallback: atomic scatter into d_out
        float* disf = (float*)(cnt + N);
        (void)hipMemsetAsync(cnt, 0, (size_t)N * 4, stream);
        (void)hipMemsetAsync(out, 0, (size_t)N * FEAT * 4, stream);
        count_kernel<<<eb, blk, 0, stream>>>(row, col, cnt, E);
        dis_kernel<<<nb, blk, 0, stream>>>(cnt, disf, N);
        const int sb = (int)(((size_t)E * 32 + 255) / 256);
        scatter_kernel<<<sb, blk, 0, stream>>>(row, col, disf, x, out, E);
        finalize_fb_kernel<<<rb, blk, 0, stream>>>(x, disf, out, N);
    }
}